// EEGGraphClf_50371376447610
// MI455X (gfx1250) — compile-verified
//
#include <hip/hip_runtime.h>
#include <hip/hip_bf16.h>
#include <math.h>

// Batched fused graph-loss:
//   loss[b] = 0.2/N^2 * (sum_n deg_n ||f_n||^2 - sum(F .* (A@F)))
//           - 0.1/N   * sum_n log(deg_n + 1e-12)
//           + 0.1/N^2 * sum(A.*A)
// A: [B,N,N] fp32, F: [B,N,D] fp32, B=64, N=1024, D=16.
// One wave per (batch, 16-row slab). A@F tile via chained V_WMMA_F32_16X16X4_F32.

typedef float v2f __attribute__((ext_vector_type(2)));
typedef float v8f __attribute__((ext_vector_type(8)));

#define N_NODES 1024
#define N_FEAT  16
#define TILES_PER_B (N_NODES / 16)   // 64 row-slabs per batch

__global__ __launch_bounds__(32)
void graph_loss_tile_kernel(const float* __restrict__ adj,
                            const float* __restrict__ feat,
                            float* __restrict__ partials)
{
    const int lane = threadIdx.x;        // 0..31, wave32
    const int n    = lane & 15;          // column-in-tile / feature index
    const int hi   = lane >> 4;          // half-wave selector (k-slots 2,3)
    const int b    = blockIdx.x >> 6;    // batch
    const int tile = blockIdx.x & 63;    // 16-row slab
    const int r0   = tile * 16;

    // Each lane owns one A row of this slab (lanes l and l^16 share a row,
    // covering complementary column subsets).
    const float* __restrict__ Arow = adj + ((size_t)b * N_NODES + (r0 + n)) * (size_t)N_NODES;
    const float* __restrict__ F    = feat + (size_t)b * N_NODES * N_FEAT;

    v8f   c   = {0.f, 0.f, 0.f, 0.f, 0.f, 0.f, 0.f, 0.f};  // AF tile accumulator
    float deg = 0.f;   // partial row-degree (this lane's columns)
    float ssq = 0.f;   // partial sum(A^2)

    // K-loop: 8 columns per iteration via one b128 load -> two WMMA (K=4 each).
    // Column->k-slot permutation: lo lanes carry cols {8s..8s+3}, hi lanes
    // {8s+4..8s+7}; B fragments use matching F rows so the dot product is exact.
    #pragma unroll 4
    for (int s = 0; s < N_NODES / 8; ++s) {
        const int k0 = 8 * s + 4 * hi;                       // 16B aligned
        const float4 q = *(const float4*)(Arow + k0);

        const float b0 = F[(k0 + 0) * N_FEAT + n];           // 64B-coalesced, WGP$-hot
        const float b1 = F[(k0 + 1) * N_FEAT + n];
        const float b2 = F[(k0 + 2) * N_FEAT + n];
        const float b3 = F[(k0 + 3) * N_FEAT + n];

        v2f aA = {q.x, q.y}, aB = {q.z, q.w};
        v2f bA = {b0, b1},   bB = {b2, b3};
        c = __builtin_amdgcn_wmma_f32_16x16x4_f32(false, aA, false, bA,
                                                  (short)0, c, false, false);
        c = __builtin_amdgcn_wmma_f32_16x16x4_f32(false, aB, false, bB,
                                                  (short)0, c, false, false);

        deg += (q.x + q.y) + (q.z + q.w);
        ssq  = fmaf(q.x, q.x, fmaf(q.y, q.y, fmaf(q.z, q.z, fmaf(q.w, q.w, ssq))));
    }

    // Full row degree: combine the two column halves held by lanes l and l^16.
    const float degFull = deg + __shfl_xor(deg, 16);

    // ||f_row||^2 for row r0+n (valid on lanes 0..15; hi lanes recompute same rows).
    const float4* __restrict__ Frow = (const float4*)(F + (size_t)(r0 + n) * N_FEAT);
    float nsq = 0.f;
    #pragma unroll
    for (int i = 0; i < 4; ++i) {
        const float4 f = Frow[i];
        nsq = fmaf(f.x, f.x, fmaf(f.y, f.y, fmaf(f.z, f.z, fmaf(f.w, f.w, nsq))));
    }

    // sum(F .* AF) over this tile from the C fragment:
    // VGPR v, lane -> AF[r0 + v + 8*hi][n]; each tile element counted once.
    float faf = 0.f;
    #pragma unroll
    for (int v = 0; v < 8; ++v) {
        const int row = r0 + v + 8 * hi;
        faf = fmaf(c[v], F[row * N_FEAT + n], faf);
    }

    const float SM_d = 0.2f / (1024.f * 1024.f);  // smoothness / N^2
    const float SP_d = 0.1f / (1024.f * 1024.f);  // sparsity   / N^2
    const float DR_n = 0.1f / 1024.f;             // degree     / N

    float part = SP_d * ssq - SM_d * faf;
    if (hi == 0) {  // degree-based terms once per row (lanes 0..15 only)
        part += SM_d * degFull * nsq - DR_n * logf(degFull + 1e-12f);
    }

    // Deterministic 32-lane tree reduction.
    #pragma unroll
    for (int off = 16; off > 0; off >>= 1)
        part += __shfl_xor(part, off);

    if (lane == 0) partials[blockIdx.x] = part;
}

__global__ __launch_bounds__(32)
void graph_loss_reduce_kernel(const float* __restrict__ partials,
                              float* __restrict__ out)
{
    const int b = blockIdx.x;
    const int l = threadIdx.x;
    float v = partials[b * TILES_PER_B + l] + partials[b * TILES_PER_B + 32 + l];
    #pragma unroll
    for (int off = 16; off > 0; off >>= 1)
        v += __shfl_xor(v, off);
    if (l == 0) out[b] = v;
}

extern "C" void kernel_launch(void* const* d_in, const int* in_sizes, int n_in,
                              void* d_out, int out_size, void* d_ws, size_t ws_size,
                              hipStream_t stream)
{
    const float* adj  = (const float*)d_in[0];   // [64,1024,1024] fp32
    const float* feat = (const float*)d_in[1];   // [64,1024,16]   fp32
    float* out        = (float*)d_out;           // [64] fp32
    float* partials   = (float*)d_ws;            // 64*64 floats = 16 KB scratch

    const int B = 64;
    graph_loss_tile_kernel<<<dim3(B * TILES_PER_B), dim3(32), 0, stream>>>(adj, feat, partials);
    graph_loss_reduce_kernel<<<dim3(B), dim3(32), 0, stream>>>(partials, out);
}